// ModelWrapper_46231027974768
// MI455X (gfx1250) — compile-verified
//
#include <hip/hip_runtime.h>
#include <hip/hip_bf16.h>

// ---------------------------------------------------------------------------
// 3-layer diagonal SSM stack for MI455X (gfx1250, wave32).
//  - chunked 3-pass parallel scan for the linear recurrence
//  - all 64x64 GEMMs via V_WMMA_F32_16X16X4_F32 (fp32 matrix path)
//  - fused bias / leaky / skip / time-downsample in GEMM epilogue
// ---------------------------------------------------------------------------

typedef __attribute__((ext_vector_type(2))) float v2f;
typedef __attribute__((ext_vector_type(8))) float v8f;

#define BATCH 16
#define T0    16000
#define T1    8000
#define T2    4000
#define NCH   64
#define NCLS  12
#define LEAK  0.01f

// scan chunking (L = chunk length, K = #chunks). L0 even so layer-0 chunk
// starts have even parity (even-t-only storage). T1=K1*L1, T2=K2*L2.
#define L0 128
#define K0 125
#define L1 125
#define K1 64
#define L2C 125
#define K2 32

// ---------------------------------------------------------------------------
// Layer-0 pass 1: fuse BU = x*B0 + Bb0 into the local chunk scan; store only
// even timesteps (downsample fused); emit per-chunk carry.
// tid = n + 64*(chunk + K0*b)
// ---------------------------------------------------------------------------
__global__ __launch_bounds__(256) void scan0_pass1(
    const float* __restrict__ x, const float* __restrict__ A,
    const float* __restrict__ Bm, const float* __restrict__ Bb,
    float* __restrict__ st0, float* __restrict__ carry)
{
    long tid = (long)blockIdx.x * blockDim.x + threadIdx.x;
    if (tid >= (long)BATCH * K0 * NCH) return;
    int  n     = (int)(tid & 63);
    long rem   = tid >> 6;
    int  chunk = (int)(rem % K0);
    int  b     = (int)(rem / K0);

    float a = A[n], bm = Bm[n], bb = Bb[n];
    const float* xp  = x   + (long)b * T0 + (long)chunk * L0;
    float*       stp = st0 + ((long)b * T1 + (long)chunk * (L0 / 2)) * NCH + n;

    float s = 0.f;
#pragma unroll 4
    for (int i = 0; i < L0; i += 2) {
        s = fmaf(s, a, fmaf(xp[i], bm, bb));       // t = chunk*L0+i (even)
        stp[(long)(i >> 1) * NCH] = s;
        s = fmaf(s, a, fmaf(xp[i + 1], bm, bb));   // odd t: not stored
    }
    carry[tid] = s;
}

// ---------------------------------------------------------------------------
// Pass 2 (all layers): serial carry scan across chunks per (b,n).
// In-place: carry[k] is replaced by c_in[k] (true state entering chunk k).
// ---------------------------------------------------------------------------
__global__ __launch_bounds__(256) void scan_pass2(
    float* carry, const float* __restrict__ A, int K, int L)
{
    int tid = blockIdx.x * blockDim.x + threadIdx.x;
    if (tid >= BATCH * NCH) return;
    int n = tid & 63;
    int b = tid >> 6;
    float a  = A[n];
    float aL = 1.f;
    for (int i = 0; i < L; ++i) aL *= a;

    float* cp = carry + n + (long)64 * (long)b * K;
    float  c  = 0.f;
    for (int k = 0; k < K; ++k) {
        float t = cp[(long)64 * k];
        cp[(long)64 * k] = c;            // c_in for chunk k
        c = fmaf(aL, c, t);
    }
}

// ---------------------------------------------------------------------------
// Layer-0 pass 3: st_even[t] += A^(i+1) * c_in, even i only => A^(2j+1).
// ---------------------------------------------------------------------------
__global__ __launch_bounds__(256) void scan0_pass3(
    float* st0, const float* __restrict__ carry, const float* __restrict__ A)
{
    long tid = (long)blockIdx.x * blockDim.x + threadIdx.x;
    if (tid >= (long)BATCH * K0 * NCH) return;
    int  n     = (int)(tid & 63);
    long rem   = tid >> 6;
    int  chunk = (int)(rem % K0);
    int  b     = (int)(rem / K0);

    float c  = carry[tid];
    float a  = A[n];
    float a2 = a * a, p = a;
    float* stp = st0 + ((long)b * T1 + (long)chunk * (L0 / 2)) * NCH + n;
#pragma unroll 4
    for (int j = 0; j < L0 / 2; ++j) {
        stp[(long)j * NCH] += p * c;
        p *= a2;
    }
}

// ---------------------------------------------------------------------------
// Generic pass 1 (layers 1,2): local scan in place (states overwrite BU).
// ---------------------------------------------------------------------------
__global__ __launch_bounds__(256) void scan_pass1g(
    float* buf, const float* __restrict__ A, float* carry,
    int Tlen, int K, int L)
{
    long tid = (long)blockIdx.x * blockDim.x + threadIdx.x;
    if (tid >= (long)BATCH * K * NCH) return;
    int  n     = (int)(tid & 63);
    long rem   = tid >> 6;
    int  chunk = (int)(rem % K);
    int  b     = (int)(rem / K);

    float a = A[n], s = 0.f;
    float* p = buf + ((long)b * Tlen + (long)chunk * L) * NCH + n;
    for (int i = 0; i < L; ++i) {
        s = fmaf(s, a, p[(long)i * NCH]);
        p[(long)i * NCH] = s;
    }
    carry[tid] = s;
}

// Generic pass 3: buf[t] += A^(i+1) * c_in.
__global__ __launch_bounds__(256) void scan_pass3g(
    float* buf, const float* __restrict__ A, const float* __restrict__ carry,
    int Tlen, int K, int L)
{
    long tid = (long)blockIdx.x * blockDim.x + threadIdx.x;
    if (tid >= (long)BATCH * K * NCH) return;
    int  n     = (int)(tid & 63);
    long rem   = tid >> 6;
    int  chunk = (int)(rem % K);
    int  b     = (int)(rem / K);

    float c  = carry[tid];
    float a  = A[n], pw = a;
    float* p = buf + ((long)b * Tlen + (long)chunk * L) * NCH + n;
    for (int i = 0; i < L; ++i) {
        p[(long)i * NCH] += pw * c;
        pw *= a;
    }
}

// ---------------------------------------------------------------------------
// out[b,t,0:64] = act( in[b, t*in_step, :] @ Wm^T + bias ) (+ skip)
// One wave32 per 16x64 output tile, V_WMMA_F32_16X16X4_F32, K=64 -> 16 slices,
// N=64 -> 4 column tiles => 64 WMMAs per wave.
// VGPR layouts (ISA 7.12.2):
//   A 16x4: v0 = {K0 | K2}, v1 = {K1 | K3} across lane halves; m = lane&15
//   B 4x16: same pattern transposed; n = lane&15
//   C/D 16x16: reg r -> rows r (lanes 0-15) / r+8 (lanes 16-31); col = lane&15
// skip_mode: 0 none; 1 buffer skip sk1[b, t*sk_step, n]; 2 x-skip
//            sk1[b, t*sk_step] * sk2[n]  (layer 0).
// ---------------------------------------------------------------------------
__global__ __launch_bounds__(256) void gemm64(
    const float* in, long T_in, int in_step,
    float* out, long T_out,
    const float* __restrict__ Wm, const float* __restrict__ bias,
    int leaky, int skip_mode,
    const float* sk1, const float* __restrict__ sk2, long sk_T, int sk_step)
{
    long wave = (long)blockIdx.x * (blockDim.x >> 5) + (threadIdx.x >> 5);
    long ntiles = (long)BATCH * T_out / 16;
    if (wave >= ntiles) return;

    long tpb = T_out >> 4;                  // 16-row tiles per batch
    int  b   = (int)(wave / tpb);
    long t0  = (wave % tpb) << 4;

    int lane = threadIdx.x & 31;
    int half = lane >> 4;
    int l15  = lane & 15;                   // A row within tile / D column

    const float* arow = in + ((long)b * T_in + (t0 + l15) * (long)in_step) * NCH;

    v8f acc[4];
#pragma unroll
    for (int nt = 0; nt < 4; ++nt) {
        float bv = bias[nt * 16 + l15];
        v8f t = {bv, bv, bv, bv, bv, bv, bv, bv};
        acc[nt] = t;
    }

#pragma unroll
    for (int ks = 0; ks < 16; ++ks) {
        int kb = ks * 4 + 2 * half;         // this lane-half's K pair
        v2f afrag = *(const v2f*)(arow + kb);
#pragma unroll
        for (int nt = 0; nt < 4; ++nt) {
            // B element (k, j) = Wm[j][k]  (out = in @ Wm^T)
            v2f bfrag = *(const v2f*)(Wm + (long)(nt * 16 + l15) * NCH + kb);
            acc[nt] = __builtin_amdgcn_wmma_f32_16x16x4_f32(
                false, afrag, false, bfrag, (short)0, acc[nt], false, false);
        }
    }

#pragma unroll
    for (int nt = 0; nt < 4; ++nt) {
#pragma unroll
        for (int r = 0; r < 8; ++r) {
            float v  = acc[nt][r];
            int  row = r + 8 * half;
            if (leaky) v = v > 0.f ? v : LEAK * v;
            if (skip_mode == 1)
                v += sk1[((long)b * sk_T + (t0 + row) * (long)sk_step) * NCH + nt * 16 + l15];
            else if (skip_mode == 2)
                v += sk1[(long)b * sk_T + (t0 + row) * (long)sk_step] * sk2[nt * 16 + l15];
            out[((long)b * T_out + t0 + row) * NCH + nt * 16 + l15] = v;
        }
    }
}

// ---------------------------------------------------------------------------
// mean over time: mean[b,n] = (1/T2) * sum_t h2[b,t,n]. Deterministic tree.
// block = (64, 4), grid = BATCH.
// ---------------------------------------------------------------------------
__global__ void mean_kernel(const float* __restrict__ h2, float* __restrict__ meanb)
{
    __shared__ float sm[4 * NCH];
    int n = threadIdx.x, q = threadIdx.y, b = blockIdx.x;
    float s = 0.f;
    for (int t = q; t < T2; t += 4)
        s += h2[((long)b * T2 + t) * NCH + n];
    sm[q * NCH + n] = s;
    __syncthreads();
    if (q == 0)
        meanb[b * NCH + n] =
            (sm[n] + sm[NCH + n] + sm[2 * NCH + n] + sm[3 * NCH + n]) * (1.f / T2);
}

// out[b,c] = mean[b,:] . W[c,:] + bias[c]
__global__ void cls_kernel(const float* __restrict__ meanb,
                           const float* __restrict__ W,
                           const float* __restrict__ bias,
                           float* __restrict__ out)
{
    int idx = threadIdx.x;
    if (idx >= BATCH * NCLS) return;
    int b = idx / NCLS, c = idx % NCLS;
    float s = bias[c];
#pragma unroll
    for (int k = 0; k < NCH; ++k)
        s += meanb[b * NCH + k] * W[c * NCH + k];
    out[idx] = s;
}

// ---------------------------------------------------------------------------
extern "C" void kernel_launch(void* const* d_in, const int* in_sizes, int n_in,
                              void* d_out, int out_size, void* d_ws, size_t ws_size,
                              hipStream_t stream)
{
    const float* x     = (const float*)d_in[0];
    const float* A0    = (const float*)d_in[1];
    const float* B0    = (const float*)d_in[2];
    const float* Bb0   = (const float*)d_in[3];
    const float* C0    = (const float*)d_in[4];
    const float* Cb0   = (const float*)d_in[5];
    const float* Skip0 = (const float*)d_in[6];
    const float* A1    = (const float*)d_in[7];
    const float* B1    = (const float*)d_in[8];
    const float* Bb1   = (const float*)d_in[9];
    const float* C1    = (const float*)d_in[10];
    const float* Cb1   = (const float*)d_in[11];
    const float* A2    = (const float*)d_in[12];
    const float* B2    = (const float*)d_in[13];
    const float* Bb2   = (const float*)d_in[14];
    const float* C2    = (const float*)d_in[15];
    const float* Cb2   = (const float*)d_in[16];
    const float* W     = (const float*)d_in[17];
    const float* bcls  = (const float*)d_in[18];

    // workspace layout (floats)
    const long szR0 = (long)BATCH * T1 * NCH;   // st0_even -> h0d (in place)
    const long szR2 = (long)BATCH * T2 * NCH;
    float* ws  = (float*)d_ws;
    float* R0  = ws;                  // 8.19M f : st0_even / h0d
    float* R1  = R0 + szR0;           // 8.19M f : BU1 -> st1 ; later h2
    float* R2  = R1 + szR0;           // 4.10M f : h1d
    float* R3  = R2 + szR2;           // 4.10M f : BU2 -> st2
    float* CAR = R3 + szR2;           // 16*64*128 carries
    float* MNB = CAR + (long)BATCH * NCH * 128; // 1024 means
    size_t need = (size_t)((MNB + BATCH * NCH) - ws) * sizeof(float);
    if (ws_size < need) return;       // insufficient scratch: refuse to corrupt

    // ---- layer 0 (T=16000 -> even-t states, T=8000) ----
    scan0_pass1<<<(BATCH * K0 * NCH + 255) / 256, 256, 0, stream>>>(x, A0, B0, Bb0, R0, CAR);
    scan_pass2 <<<(BATCH * NCH + 255) / 256,      256, 0, stream>>>(CAR, A0, K0, L0);
    scan0_pass3<<<(BATCH * K0 * NCH + 255) / 256, 256, 0, stream>>>(R0, CAR, A0);
    // h0d = leaky(st0_even @ C0^T + Cb0) + x[:, ::2]*Skip0   (in place)
    gemm64<<<(T1 + 7) / 8, 256, 0, stream>>>(R0, T1, 1, R0, T1, C0, Cb0,
                                             1, 2, x, Skip0, T0, 2);

    // ---- layer 1 (T=8000) ----
    gemm64<<<(T1 + 7) / 8, 256, 0, stream>>>(R0, T1, 1, R1, T1, B1, Bb1,
                                             0, 0, nullptr, nullptr, 0, 0);
    scan_pass1g<<<(BATCH * K1 * NCH + 255) / 256, 256, 0, stream>>>(R1, A1, CAR, T1, K1, L1);
    scan_pass2 <<<(BATCH * NCH + 255) / 256,      256, 0, stream>>>(CAR, A1, K1, L1);
    scan_pass3g<<<(BATCH * K1 * NCH + 255) / 256, 256, 0, stream>>>(R1, A1, CAR, T1, K1, L1);
    // h1d = leaky(st1[::2] @ C1^T + Cb1) + h0d[::2]
    gemm64<<<(T2 + 7) / 8, 256, 0, stream>>>(R1, T1, 2, R2, T2, C1, Cb1,
                                             1, 1, R0, nullptr, T1, 2);

    // ---- layer 2 (T=4000) ----
    gemm64<<<(T2 + 7) / 8, 256, 0, stream>>>(R2, T2, 1, R3, T2, B2, Bb2,
                                             0, 0, nullptr, nullptr, 0, 0);
    scan_pass1g<<<(BATCH * K2 * NCH + 255) / 256, 256, 0, stream>>>(R3, A2, CAR, T2, K2, L2C);
    scan_pass2 <<<(BATCH * NCH + 255) / 256,      256, 0, stream>>>(CAR, A2, K2, L2C);
    scan_pass3g<<<(BATCH * K2 * NCH + 255) / 256, 256, 0, stream>>>(R3, A2, CAR, T2, K2, L2C);
    // h2 = leaky(st2 @ C2^T + Cb2) + h1d  -> R1 (reused)
    gemm64<<<(T2 + 7) / 8, 256, 0, stream>>>(R3, T2, 1, R1, T2, C2, Cb2,
                                             1, 1, R2, nullptr, T2, 1);

    // ---- head ----
    mean_kernel<<<BATCH, dim3(NCH, 4), 0, stream>>>(R1, MNB);
    cls_kernel<<<1, 256, 0, stream>>>(MNB, W, bcls, (float*)d_out);
}